// BennaSynapse_71717363909113
// MI455X (gfx1250) — compile-verified
//
#include <hip/hip_runtime.h>
#include <math.h>

#define D0 2048
#define D1 2048
#define NCLS 47
#define NJS 16            // j-reduction slices per matvec (128 rows each)

typedef __attribute__((ext_vector_type(2))) float v2f;
typedef __attribute__((ext_vector_type(8))) float v8f;

// d_ws layout (floats)
#define WS_E1  0
#define WS_E0  2048
#define WS_V   4096
#define WS_SA1 6144
#define WS_S6  6145
#define WS_T   6146

struct BennaConst { float G[6]; float invC[5]; };

// ---------------- K1: softmax -> e2 -> e1, Sa1, zero accumulators ----------------
__global__ void k1_prep(const float* __restrict__ logits,
                        const int*   __restrict__ label,
                        const float* __restrict__ f2,
                        const float* __restrict__ a1,
                        const int*   __restrict__ beta_p,
                        float* __restrict__ ws) {
  __shared__ float e2s[NCLS];
  __shared__ float red[256];
  const int t = threadIdx.x;
  const float beta = (float)(*beta_p);
  if (t == 0) {
    const int lab = label[0];
    float m = logits[0];
    for (int c = 1; c < NCLS; ++c) m = fmaxf(m, logits[c]);
    float s = 0.f;
    for (int c = 0; c < NCLS; ++c) s += expf(logits[c] - m);
    const float inv = 1.f / s;
    for (int c = 0; c < NCLS; ++c)
      e2s[c] = expf(logits[c] - m) * inv - (c == lab ? 1.f : 0.f);
    ws[WS_T] = 0.f;  // reset atomic scalar accumulator each call
  }
  __syncthreads();
  // zero e0 and v atomic-accumulation buffers (contiguous: 2048..6143)
  for (int i = t; i < 2 * D0; i += 256) ws[WS_E0 + i] = 0.f;
  // e1[j] = (e2 @ f2)[j] * (1 - exp(-beta*a1[j]))
  for (int j = t; j < D1; j += 256) {
    float acc = 0.f;
    for (int c = 0; c < NCLS; ++c) acc += e2s[c] * f2[c * D1 + j];
    ws[WS_E1 + j] = acc * (1.f - expf(-beta * a1[j]));
  }
  // Sa1 = sum(a1)
  float s = 0.f;
  for (int j = t; j < D1; j += 256) s += a1[j];
  red[t] = s; __syncthreads();
  for (int off = 128; off; off >>= 1) {
    if (t < off) red[t] += red[t + off];
    __syncthreads();
  }
  if (t == 0) ws[WS_SA1] = red[0];
}

// ---- y[i] += sum_{j in slice} x[j]*W[j,i] via WMMA f32 16x16x4, 32-col tile ----
// A[m,k] = W[j+k, i0+m] (coalesced 64B row segments), B[k,n] = x[j+k] (const over n)
// => D[m,n] accumulates y[i0+m] replicated over n.
// Optional fusion: t_acc += sum_{j,i covered} rw[j]*W[j,i]*cw[i] (per-lane from A regs).
__global__ void k_matvec_wmma(const float* __restrict__ W,   // (D1, D0) row-major
                              const float* __restrict__ x,   // (D1)
                              const float* __restrict__ rw,  // row weights (e1) or null
                              const float* __restrict__ cw,  // col weights (a0) or null
                              float* __restrict__ y,         // (D0), pre-zeroed
                              float* t_acc) {                // scalar accumulator or null
  const int lane = threadIdx.x;      // blockDim.x == 32 (one wave)
  const int half = lane >> 4;
  const int li   = lane & 15;
  const int i0   = blockIdx.x * 32;
  const int jb   = blockIdx.y * (D1 / NJS);

  v8f cA0 = {}, cA1 = {}, cB0 = {}, cB1 = {};
  float tp = 0.f, wc0 = 0.f, wc1 = 0.f;
  if (t_acc) { wc0 = cw[i0 + li]; wc1 = cw[i0 + 16 + li]; }

  for (int jj = jb; jj < jb + (D1 / NJS); jj += 16) {
    const int r0 = jj + 2 * half;    // this lane's K rows for chunk 0
    const float* p = W + (size_t)r0 * D0 + i0 + li;
    v2f aA0, aA1, aA2, aA3, aB0, aB1, aB2, aB3, b0, b1, b2, b3;
    aA0.x = p[0];           aA0.y = p[D0];
    aA1.x = p[4 * D0];      aA1.y = p[5 * D0];
    aA2.x = p[8 * D0];      aA2.y = p[9 * D0];
    aA3.x = p[12 * D0];     aA3.y = p[13 * D0];
    aB0.x = p[16];          aB0.y = p[D0 + 16];
    aB1.x = p[4 * D0 + 16]; aB1.y = p[5 * D0 + 16];
    aB2.x = p[8 * D0 + 16]; aB2.y = p[9 * D0 + 16];
    aB3.x = p[12 * D0 + 16]; aB3.y = p[13 * D0 + 16];
    b0.x = x[r0];      b0.y = x[r0 + 1];
    b1.x = x[r0 + 4];  b1.y = x[r0 + 5];
    b2.x = x[r0 + 8];  b2.y = x[r0 + 9];
    b3.x = x[r0 + 12]; b3.y = x[r0 + 13];
    cA0 = __builtin_amdgcn_wmma_f32_16x16x4_f32(false, aA0, false, b0, (short)0, cA0, false, false);
    cB0 = __builtin_amdgcn_wmma_f32_16x16x4_f32(false, aB0, false, b0, (short)0, cB0, false, false);
    cA1 = __builtin_amdgcn_wmma_f32_16x16x4_f32(false, aA1, false, b1, (short)0, cA1, false, false);
    cB1 = __builtin_amdgcn_wmma_f32_16x16x4_f32(false, aB1, false, b1, (short)0, cB1, false, false);
    cA0 = __builtin_amdgcn_wmma_f32_16x16x4_f32(false, aA2, false, b2, (short)0, cA0, false, false);
    cB0 = __builtin_amdgcn_wmma_f32_16x16x4_f32(false, aB2, false, b2, (short)0, cB0, false, false);
    cA1 = __builtin_amdgcn_wmma_f32_16x16x4_f32(false, aA3, false, b3, (short)0, cA1, false, false);
    cB1 = __builtin_amdgcn_wmma_f32_16x16x4_f32(false, aB3, false, b3, (short)0, cB1, false, false);
    if (t_acc) {
      const float w0 = rw[r0],      w1v = rw[r0 + 1];
      const float w2 = rw[r0 + 4],  w3 = rw[r0 + 5];
      const float w4 = rw[r0 + 8],  w5 = rw[r0 + 9];
      const float w6 = rw[r0 + 12], w7 = rw[r0 + 13];
      const float sA = aA0.x * w0 + aA0.y * w1v + aA1.x * w2 + aA1.y * w3
                     + aA2.x * w4 + aA2.y * w5 + aA3.x * w6 + aA3.y * w7;
      const float sB = aB0.x * w0 + aB0.y * w1v + aB1.x * w2 + aB1.y * w3
                     + aB2.x * w4 + aB2.y * w5 + aB3.x * w6 + aB3.y * w7;
      tp += sA * wc0 + sB * wc1;
    }
  }
  v8f cA = cA0 + cA1, cB = cB0 + cB1;
  // D layout: lanes 0-15 vgpr r -> M=r; lanes 16-31 vgpr r -> M=r+8; all N equal.
  if (li == 0) {
    const int base = i0 + half * 8;
#pragma unroll
    for (int r = 0; r < 8; ++r) {
      atomicAdd(&y[base + r], cA[r]);
      atomicAdd(&y[base + 16 + r], cB[r]);
    }
  }
  if (t_acc) {
#pragma unroll
    for (int off = 16; off; off >>= 1) tp += __shfl_xor(tp, off, 32);
    if (lane == 0) atomicAdd(t_acc, tp);
  }
}

// ------------- tiny pass: apply gate to e0 after atomic accumulation -------------
__global__ void k_gate_e0(const float* __restrict__ a0,
                          const int* __restrict__ beta_p,
                          float* ws) {
  const int i = blockIdx.x * blockDim.x + threadIdx.x;
  const float beta = (float)(*beta_p);
  ws[WS_E0 + i] *= (1.f - expf(-beta * a0[i]));
}

// ------------- s6 = v . e0 -------------
__global__ void k_dot_s6(float* ws) {
  __shared__ float red[256];
  const int t = threadIdx.x;
  float s = 0.f;
  for (int i = t; i < D0; i += 256) s += ws[WS_V + i] * ws[WS_E0 + i];
  red[t] = s; __syncthreads();
  for (int off = 128; off; off >>= 1) {
    if (t < off) red[t] += red[t + off];
    __syncthreads();
  }
  if (t == 0) ws[WS_S6] = red[0];
}

// ------------- K4: fused tanh(inChange[0]) + Benna-Fusi diffusion -------------
__global__ void k_main(const float* __restrict__ w1,
                       const float* __restrict__ a0,
                       const float* __restrict__ a1,
                       const float* __restrict__ h1,
                       const float* __restrict__ P,
                       const float* __restrict__ ws,
                       float* __restrict__ out,
                       BennaConst bc) {
  __shared__ float cf[4];
  const int j = blockIdx.x;
  const int t = threadIdx.x;
  if (t == 0) {
    const float e1j = ws[WS_E1 + j];
    const float a1j = a1[j];
    const float Sa1 = ws[WS_SA1], s6 = ws[WS_S6], tt = ws[WS_T];
    // only P row 0 feeds the dynamics (inChange[0] is the only channel used)
    const float P0 = P[0], P1 = P[1], P2 = P[2], P3 = P[3], P4 = P[4],
                P5 = P[5], P6 = P[6], P7 = P[7], P8 = P[8], P9 = P[9];
    cf[0] = -(P0 + P5 * Sa1 + P7 * s6) * e1j + P9 * a1j;       // * a0[i]
    cf[1] = -(P1 + P6 * s6 + P8 * tt) * a1j - P2 * e1j - P4;   // * e0[i]
    cf[2] = -P9 * a1j;                                          // * v[i]
    cf[3] = -P3;                                                // * w1[j,i]
  }
  __syncthreads();
  const float ca = cf[0], ce = cf[1], cv = cf[2], cw = cf[3];
  const size_t row = (size_t)j * D0;
  const size_t plane = (size_t)D1 * D0;

  auto elem = [&](float wi, float a0i, float e0i, float vi,
                  float h0, float h1v, float h2, float h3, float h4,
                  float* o) {
    const float ich = tanhf(fmaf(ca, a0i, fmaf(ce, e0i, fmaf(cv, vi, cw * wi))));
    const float f0 = bc.G[0] * (h0 - h1v);
    const float f1 = bc.G[1] * (h1v - h2);
    const float f2 = bc.G[2] * (h2 - h3);
    const float f3 = bc.G[3] * (h3 - h4);
    const float b0 = bc.G[1] * (h1v - h0);
    const float b1 = bc.G[2] * (h2 - h1v);
    const float b2 = bc.G[3] * (h3 - h2);
    const float b3 = bc.G[4] * (h4 - h3);
    o[0] = h0  + (ich + b0) * bc.invC[0];
    o[1] = h1v + (f0 + b1) * bc.invC[1];
    o[2] = h2  + (f1 + b2) * bc.invC[2];
    o[3] = h3  + (f2 + b3) * bc.invC[3];
    o[4] = h4  + (f3 - bc.G[5] * h4) * bc.invC[4];
  };

#pragma unroll
  for (int q = 0; q < 2; ++q) {
    const int i = t * 8 + q * 4;
    const float4 wv = *(const float4*)(w1 + row + i);
    const float4 av = *(const float4*)(a0 + i);
    const float4 ev = *(const float4*)(ws + WS_E0 + i);
    const float4 vv = *(const float4*)(ws + WS_V + i);
    float4 h[5];
#pragma unroll
    for (int c = 0; c < 5; ++c) h[c] = *(const float4*)(h1 + c * plane + row + i);
    float4 o[5];
    float tmp[5];
    elem(wv.x, av.x, ev.x, vv.x, h[0].x, h[1].x, h[2].x, h[3].x, h[4].x, tmp);
    o[0].x = tmp[0]; o[1].x = tmp[1]; o[2].x = tmp[2]; o[3].x = tmp[3]; o[4].x = tmp[4];
    elem(wv.y, av.y, ev.y, vv.y, h[0].y, h[1].y, h[2].y, h[3].y, h[4].y, tmp);
    o[0].y = tmp[0]; o[1].y = tmp[1]; o[2].y = tmp[2]; o[3].y = tmp[3]; o[4].y = tmp[4];
    elem(wv.z, av.z, ev.z, vv.z, h[0].z, h[1].z, h[2].z, h[3].z, h[4].z, tmp);
    o[0].z = tmp[0]; o[1].z = tmp[1]; o[2].z = tmp[2]; o[3].z = tmp[3]; o[4].z = tmp[4];
    elem(wv.w, av.w, ev.w, vv.w, h[0].w, h[1].w, h[2].w, h[3].w, h[4].w, tmp);
    o[0].w = tmp[0]; o[1].w = tmp[1]; o[2].w = tmp[2]; o[3].w = tmp[3]; o[4].w = tmp[4];
#pragma unroll
    for (int c = 0; c < 5; ++c) *(float4*)(out + c * plane + row + i) = o[c];
  }
}

extern "C" void kernel_launch(void* const* d_in, const int* in_sizes, int n_in,
                              void* d_out, int out_size, void* d_ws, size_t ws_size,
                              hipStream_t stream) {
  (void)in_sizes; (void)n_in; (void)out_size; (void)ws_size;
  const float* a0     = (const float*)d_in[0];
  const float* a1     = (const float*)d_in[1];
  const float* logits = (const float*)d_in[2];
  const int*   label  = (const int*)d_in[3];
  const float* w1     = (const float*)d_in[4];
  const float* f1     = (const float*)d_in[5];
  const float* f2     = (const float*)d_in[6];
  const float* h1     = (const float*)d_in[7];
  const float* P      = (const float*)d_in[8];
  const int*   beta   = (const int*)d_in[9];
  float* ws  = (float*)d_ws;
  float* out = (float*)d_out;

  // Benna constants: tau = 50^(1/8); C_vec = tau^k; G_vec = tau^k (MIN_TAU=1)
  BennaConst bc;
  const double tau = pow(50.0, 1.0 / 8.0);
  for (int k = 0; k < 6; ++k) bc.G[k] = (float)pow(tau, (double)k);
  for (int k = 0; k < 5; ++k) bc.invC[k] = (float)(1.0 / pow(tau, (double)k));

  const dim3 mg(D0 / 32, NJS);

  // K1: e2, e1, Sa1; zero e0/v/t accumulators
  k1_prep<<<1, 256, 0, stream>>>(logits, label, f2, a1, beta, ws);
  // K2: e0_raw += slices of e1 @ f1   [WMMA, 1024 waves]
  k_matvec_wmma<<<mg, 32, 0, stream>>>(f1, ws + WS_E1, nullptr, nullptr,
                                       ws + WS_E0, nullptr);
  // gate: e0 *= (1 - exp(-beta*a0))
  k_gate_e0<<<D0 / 256, 256, 0, stream>>>(a0, beta, ws);
  // K3: v += slices of a1 @ w1; fused t += e1.(w1 rows).a0   [WMMA, 1024 waves]
  k_matvec_wmma<<<mg, 32, 0, stream>>>(w1, a1, ws + WS_E1, a0,
                                       ws + WS_V, ws + WS_T);
  // s6 = v . e0
  k_dot_s6<<<1, 256, 0, stream>>>(ws);
  // K4: fused inChange[0] + Benna update over (5, 2048, 2048)
  k_main<<<D1, 256, 0, stream>>>(w1, a0, a1, h1, P, ws, out, bc);
}